// GATclf_63788854280506
// MI455X (gfx1250) — compile-verified
//
#include <hip/hip_runtime.h>
#include <math.h>

// ---------------------------------------------------------------------------
// GAT classifier pipeline for gfx1250 (MI455X).
// Roofline: dense GEMMs ~36 GFLOP (trivial on the WMMA pipe), edge
// gather/scatter ~2-3 GB of irregular traffic -> memory/atomic bound.
// GEMMs use full-precision f32 WMMA (v_wmma_f32_16x16x4_f32) with 1x4
// column-tile register blocking: one A-fragment b64 load feeds 4 WMMAs,
// A re-read factor is Fout/64 (fits 192MB L2 for every layer).
// ---------------------------------------------------------------------------

typedef __attribute__((ext_vector_type(2))) float v2f;
typedef __attribute__((ext_vector_type(8))) float v8f;

#define NEG_SLOPE 0.2f
#define BN_EPS 1e-5f

// ---------------- WMMA GEMM: C[N,Fout] = A[N,K] * W[K,Fout] (+bias)(+relu) --
// One wave per 16x64 output slab (4 x 16x16 WMMA tiles). N mult of 16,
// Fout mult of 64, K mult of 4.
// A fragment (16x4 f32, ISA 7.12.2): lanes 0-15 hold K={0,1}, lanes 16-31
// K={2,3}, row M = lane&15 -> one contiguous 8B load per K-step.
// B fragment mirrored with N = lane&15. D: M = (lane>=16)*8 + i, N = lane&15.
__global__ __launch_bounds__(256) void gemm_wmma_f32(
    const float* __restrict__ A, const float* __restrict__ W,
    const float* __restrict__ bias, float* __restrict__ C,
    int N, int K, int Fout, int act) {
  const int lane = threadIdx.x & 31;
  const int wave = threadIdx.x >> 5;
  const int cgroups = Fout >> 6;                       // 64-col groups
  const int slab = blockIdx.x * (blockDim.x >> 5) + wave;
  const int total = (N >> 4) * cgroups;
  if (slab >= total) return;           // uniform per-wave: EXEC stays all-1s
  const int tm = slab / cgroups;
  const int cg = slab - tm * cgroups;
  const int row = tm * 16 + (lane & 15);               // A row (M)
  const int col0 = cg * 64 + (lane & 15);              // first B col (N)
  const int khalf = (lane >> 4) * 2;                   // 0 / 2

  v8f acc0 = {}, acc1 = {}, acc2 = {}, acc3 = {};
  const float* arow = A + (size_t)row * K;
#pragma unroll 2
  for (int k = 0; k < K; k += 4) {
    const v2f a = *(const v2f*)(arow + k + khalf);     // K={kh,kh+1}, 8B load
    const float* w0 = W + (size_t)(k + khalf) * Fout;
    const float* w1 = w0 + Fout;
    v2f b0, b1, b2, b3;
    b0.x = w0[col0];      b0.y = w1[col0];
    b1.x = w0[col0 + 16]; b1.y = w1[col0 + 16];
    b2.x = w0[col0 + 32]; b2.y = w1[col0 + 32];
    b3.x = w0[col0 + 48]; b3.y = w1[col0 + 48];
    acc0 = __builtin_amdgcn_wmma_f32_16x16x4_f32(false, a, false, b0,
                                                 (short)0, acc0, false, false);
    acc1 = __builtin_amdgcn_wmma_f32_16x16x4_f32(false, a, false, b1,
                                                 (short)0, acc1, false, false);
    acc2 = __builtin_amdgcn_wmma_f32_16x16x4_f32(false, a, false, b2,
                                                 (short)0, acc2, false, false);
    acc3 = __builtin_amdgcn_wmma_f32_16x16x4_f32(false, a, false, b3,
                                                 (short)0, acc3, false, false);
  }
  const int mbase = tm * 16 + (lane >> 4) * 8;
  float bv0 = 0.f, bv1 = 0.f, bv2 = 0.f, bv3 = 0.f;
  if (bias) {
    bv0 = bias[col0];      bv1 = bias[col0 + 16];
    bv2 = bias[col0 + 32]; bv3 = bias[col0 + 48];
  }
#pragma unroll
  for (int i = 0; i < 8; ++i) {
    float* crow = C + (size_t)(mbase + i) * Fout;
    float v0 = acc0[i] + bv0, v1 = acc1[i] + bv1;
    float v2 = acc2[i] + bv2, v3 = acc3[i] + bv3;
    if (act) {
      v0 = fmaxf(v0, 0.f); v1 = fmaxf(v1, 0.f);
      v2 = fmaxf(v2, 0.f); v3 = fmaxf(v3, 0.f);
    }
    crow[col0]      = v0;
    crow[col0 + 16] = v1;
    crow[col0 + 32] = v2;
    crow[col0 + 48] = v3;
  }
}

// ---------------- attention dot products: a_s[n]=h[n]·a_src, a_d[n]=h[n]·a_dst
__global__ __launch_bounds__(256) void attn_dots(
    const float* __restrict__ h, const float* __restrict__ asrc,
    const float* __restrict__ adst, float* __restrict__ as_out,
    float* __restrict__ ad_out, int N, int F) {
  const int gw = (int)((blockIdx.x * blockDim.x + threadIdx.x) >> 5);
  const int lane = threadIdx.x & 31;
  if (gw >= N) return;
  float s = 0.0f, d = 0.0f;
  const float* hr = h + (size_t)gw * F;
  for (int f = lane; f < F; f += 32) {
    const float hv = hr[f];
    s += hv * asrc[f];
    d += hv * adst[f];
  }
  for (int off = 16; off; off >>= 1) {
    s += __shfl_down(s, off, 32);
    d += __shfl_down(d, off, 32);
  }
  if (lane == 0) { as_out[gw] = s; ad_out[gw] = d; }
}

// monotone float<->uint map for atomicMax on signed floats
__device__ __forceinline__ unsigned f2mono(float f) {
  unsigned b = __float_as_uint(f);
  return (b & 0x80000000u) ? ~b : (b | 0x80000000u);
}
__device__ __forceinline__ float mono2f(unsigned u) {
  return (u & 0x80000000u) ? __uint_as_float(u & 0x7FFFFFFFu)
                           : __uint_as_float(~u);
}

__global__ void init_node_accum(unsigned* __restrict__ m, float* __restrict__ z, int N) {
  const int i = blockIdx.x * blockDim.x + threadIdx.x;
  if (i < N) { m[i] = 0x007FFFFFu; /* mono(-inf) */ z[i] = 0.0f; }
}

__global__ void init_bias_rows(float* __restrict__ out, const float* __restrict__ b,
                               int N, int F) {
  const int i = blockIdx.x * blockDim.x + threadIdx.x;
  if (i < N * F) out[i] = b[i % F];
}

__global__ void zero_f32(float* __restrict__ p, int n) {
  const int i = blockIdx.x * blockDim.x + threadIdx.x;
  if (i < n) p[i] = 0.0f;
}

// pass 1: e = leaky_relu(a_s[src]+a_d[dst]); segment max over dst
__global__ void edge_max(const int* __restrict__ src, const int* __restrict__ dst,
                         const float* __restrict__ as_in, const float* __restrict__ ad_in,
                         unsigned* __restrict__ m, float* __restrict__ e_out,
                         int E, int N) {
  const int i = blockIdx.x * blockDim.x + threadIdx.x;
  if (i >= E + N) return;
  const int s = (i < E) ? src[i] : (i - E);
  const int d = (i < E) ? dst[i] : (i - E);
  float e = as_in[s] + ad_in[d];
  e = (e > 0.0f) ? e : NEG_SLOPE * e;
  e_out[i] = e;
  atomicMax(&m[d], f2mono(e));
}

// pass 2: ex = exp(e - max[dst]); segment sum over dst
__global__ void edge_exp(const int* __restrict__ dst, const unsigned* __restrict__ m,
                         float* __restrict__ e_inout, float* __restrict__ z,
                         int E, int N) {
  const int i = blockIdx.x * blockDim.x + threadIdx.x;
  if (i >= E + N) return;
  const int d = (i < E) ? dst[i] : (i - E);
  const float ex = expf(e_inout[i] - mono2f(m[d]));
  e_inout[i] = ex;
  atomicAdd(&z[d], ex);
}

// pass 3: out[dst] += (ex/z[dst]) * h[src] ; one wave per edge, lanes = features
__global__ __launch_bounds__(256) void edge_aggr(
    const int* __restrict__ src, const int* __restrict__ dst,
    const float* __restrict__ ex, const float* __restrict__ z,
    const float* __restrict__ h, float* __restrict__ out, int E, int N, int F) {
  const int gw = (int)((blockIdx.x * blockDim.x + threadIdx.x) >> 5);
  const int lane = threadIdx.x & 31;
  if (gw >= E + N) return;
  const int s = (gw < E) ? src[gw] : (gw - E);
  const int d = (gw < E) ? dst[gw] : (gw - E);
  const float alpha = ex[gw] / z[d];
  const float* hr = h + (size_t)s * F;
  float* orow = out + (size_t)d * F;
  for (int f = lane; f < F; f += 32) atomicAdd(&orow[f], alpha * hr[f]);
}

// ---------------- batchnorm statistics (sum, sumsq per feature) -------------
__global__ __launch_bounds__(256) void bn_stats(const float* __restrict__ x,
                                                float* __restrict__ sum,
                                                float* __restrict__ sumsq,
                                                int N, int F) {
  __shared__ float ls[256];
  __shared__ float lq[256];
  for (int f = threadIdx.x; f < F; f += blockDim.x) { ls[f] = 0.0f; lq[f] = 0.0f; }
  __syncthreads();
  const int total = N * F;
  const int stride = gridDim.x * blockDim.x;
  for (int idx = blockIdx.x * blockDim.x + threadIdx.x; idx < total; idx += stride) {
    const float v = x[idx];
    const int f = idx % F;
    atomicAdd(&ls[f], v);
    atomicAdd(&lq[f], v * v);
  }
  __syncthreads();
  for (int f = threadIdx.x; f < F; f += blockDim.x) {
    atomicAdd(&sum[f], ls[f]);
    atomicAdd(&sumsq[f], lq[f]);
  }
}

__global__ void bn_apply_relu(float* __restrict__ x, const float* __restrict__ sum,
                              const float* __restrict__ sumsq,
                              const float* __restrict__ gamma,
                              const float* __restrict__ beta, int N, int F) {
  const int idx = blockIdx.x * blockDim.x + threadIdx.x;
  if (idx >= N * F) return;
  const int f = idx % F;
  const float invN = 1.0f / (float)N;
  const float mu = sum[f] * invN;
  const float var = sumsq[f] * invN - mu * mu;
  const float v = (x[idx] - mu) * rsqrtf(var + BN_EPS) * gamma[f] + beta[f];
  x[idx] = fmaxf(v, 0.0f);
}

// ---------------- final 128->1 dot + sigmoid --------------------------------
__global__ __launch_bounds__(256) void final_out_sigmoid(
    const float* __restrict__ x, const float* __restrict__ W,
    const float* __restrict__ b, float* __restrict__ out, int N, int F) {
  const int gw = (int)((blockIdx.x * blockDim.x + threadIdx.x) >> 5);
  const int lane = threadIdx.x & 31;
  if (gw >= N) return;
  float s = 0.0f;
  const float* xr = x + (size_t)gw * F;
  for (int f = lane; f < F; f += 32) s += xr[f] * W[f];
  for (int off = 16; off; off >>= 1) s += __shfl_down(s, off, 32);
  if (lane == 0) out[gw] = 1.0f / (1.0f + expf(-(s + b[0])));
}

// ---------------------------------------------------------------------------
static inline void launch_gemm(const float* A, const float* W, const float* bias,
                               float* C, int N, int K, int Fout, int act,
                               hipStream_t s) {
  const int slabs = (N >> 4) * (Fout >> 6);  // 16x64 slabs, Fout mult of 64
  const int blocks = (slabs + 7) / 8;        // 8 waves per 256-thread block
  hipLaunchKernelGGL(gemm_wmma_f32, dim3(blocks), dim3(256), 0, s,
                     A, W, bias, C, N, K, Fout, act);
}

extern "C" void kernel_launch(void* const* d_in, const int* in_sizes, int n_in,
                              void* d_out, int out_size, void* d_ws, size_t ws_size,
                              hipStream_t stream) {
  const int N = in_sizes[0] / 32;          // 50000
  const int E = in_sizes[1] / 2;           // 1600000
  const int ET = E + N;                    // edges + self loops

  const float* x0 = (const float*)d_in[0];
  const int* src = (const int*)d_in[1];
  const int* dst = src + E;

  // encoder (W,b) x4 : indices 2..9   dims 32->128->256->128->128
  const float* encW[4] = {(const float*)d_in[2], (const float*)d_in[4],
                          (const float*)d_in[6], (const float*)d_in[8]};
  const float* encB[4] = {(const float*)d_in[3], (const float*)d_in[5],
                          (const float*)d_in[7], (const float*)d_in[9]};
  // gat[i]: W, a_src, a_dst, b at 10+4i
  const float *gatW[3], *gatAS[3], *gatAD[3], *gatB[3];
  for (int i = 0; i < 3; ++i) {
    gatW[i]  = (const float*)d_in[10 + 4 * i];
    gatAS[i] = (const float*)d_in[11 + 4 * i];
    gatAD[i] = (const float*)d_in[12 + 4 * i];
    gatB[i]  = (const float*)d_in[13 + 4 * i];
  }
  // bn[i]: gamma, beta at 22+2i
  const float *bnG[3], *bnB[3];
  for (int i = 0; i < 3; ++i) {
    bnG[i] = (const float*)d_in[22 + 2 * i];
    bnB[i] = (const float*)d_in[23 + 2 * i];
  }
  // nup[0]: 28..35, nup[1]: 36..43, out_mlp: 44..51 (each 4 (W,b) pairs)
  const float *n0W[4], *n0B[4], *n1W[4], *n1B[4], *omW[4], *omB[4];
  for (int i = 0; i < 4; ++i) {
    n0W[i] = (const float*)d_in[28 + 2 * i];  n0B[i] = (const float*)d_in[29 + 2 * i];
    n1W[i] = (const float*)d_in[36 + 2 * i];  n1B[i] = (const float*)d_in[37 + 2 * i];
    omW[i] = (const float*)d_in[44 + 2 * i];  omB[i] = (const float*)d_in[45 + 2 * i];
  }

  // workspace carve-up (floats)
  float* ws = (float*)d_ws;
  float* bufA = ws;                      // N*256
  float* bufB = bufA + (size_t)N * 256;  // N*256
  float* bufH = bufB + (size_t)N * 256;  // N*256
  float* a_s  = bufH + (size_t)N * 256;  // N
  float* a_d  = a_s + N;                 // N
  unsigned* mx = (unsigned*)(a_d + N);   // N
  float* z    = (float*)(mx + N);        // N
  float* ex   = z + N;                   // E+N
  float* bnSum = ex + ET;                // 256
  float* bnSq  = bnSum + 256;            // 256

  const int T = 256;
  auto cdiv = [](int a, int b) { return (a + b - 1) / b; };

  // ---- encoder MLP: 32 ->128 ->256 ->128 ->128 (relu on first 3) ----
  launch_gemm(x0,   encW[0], encB[0], bufA, N, 32, 128, 1, stream);
  launch_gemm(bufA, encW[1], encB[1], bufB, N, 128, 256, 1, stream);
  launch_gemm(bufB, encW[2], encB[2], bufA, N, 256, 128, 1, stream);
  launch_gemm(bufA, encW[3], encB[3], bufB, N, 128, 128, 0, stream);
  // x in bufB, F=128

  auto gat_layer = [&](const float* xin, int Fin, int Fout, int g, float* out) {
    launch_gemm(xin, gatW[g], nullptr, bufH, N, Fin, Fout, 0, stream);  // h
    hipLaunchKernelGGL(attn_dots, dim3(cdiv(N, 8)), dim3(T), 0, stream,
                       bufH, gatAS[g], gatAD[g], a_s, a_d, N, Fout);
    hipLaunchKernelGGL(init_node_accum, dim3(cdiv(N, T)), dim3(T), 0, stream, mx, z, N);
    hipLaunchKernelGGL(edge_max, dim3(cdiv(ET, T)), dim3(T), 0, stream,
                       src, dst, a_s, a_d, mx, ex, E, N);
    hipLaunchKernelGGL(edge_exp, dim3(cdiv(ET, T)), dim3(T), 0, stream,
                       dst, mx, ex, z, E, N);
    hipLaunchKernelGGL(init_bias_rows, dim3(cdiv(N * Fout, T)), dim3(T), 0, stream,
                       out, gatB[g], N, Fout);
    hipLaunchKernelGGL(edge_aggr, dim3(cdiv(ET, 8)), dim3(T), 0, stream,
                       src, dst, ex, z, bufH, out, E, N, Fout);
  };

  auto batchnorm_relu = [&](float* xb, int F, int g) {
    hipLaunchKernelGGL(zero_f32, dim3(2), dim3(T), 0, stream, bnSum, 512);
    hipLaunchKernelGGL(bn_stats, dim3(512), dim3(T), 0, stream, xb, bnSum, bnSq, N, F);
    hipLaunchKernelGGL(bn_apply_relu, dim3(cdiv(N * F, T)), dim3(T), 0, stream,
                       xb, bnSum, bnSq, bnG[g], bnB[g], N, F);
  };

  // ---- step 0: GAT 128->64, bn, relu, nup0 (64->128->256->128->64, all relu)
  gat_layer(bufB, 128, 64, 0, bufA);
  batchnorm_relu(bufA, 64, 0);
  launch_gemm(bufA, n0W[0], n0B[0], bufB, N, 64, 128, 1, stream);
  launch_gemm(bufB, n0W[1], n0B[1], bufA, N, 128, 256, 1, stream);
  launch_gemm(bufA, n0W[2], n0B[2], bufB, N, 256, 128, 1, stream);
  launch_gemm(bufB, n0W[3], n0B[3], bufA, N, 128, 64, 1, stream);
  // x in bufA, F=64

  // ---- step 1: GAT 64->128, bn, relu, nup1 (128->128->256->128->128)
  gat_layer(bufA, 64, 128, 1, bufB);
  batchnorm_relu(bufB, 128, 1);
  launch_gemm(bufB, n1W[0], n1B[0], bufA, N, 128, 128, 1, stream);
  launch_gemm(bufA, n1W[1], n1B[1], bufB, N, 128, 256, 1, stream);
  launch_gemm(bufB, n1W[2], n1B[2], bufA, N, 256, 128, 1, stream);
  launch_gemm(bufA, n1W[3], n1B[3], bufB, N, 128, 128, 1, stream);
  // x in bufB, F=128

  // ---- step 2: final GAT 128->64, bn, relu
  gat_layer(bufB, 128, 64, 2, bufA);
  batchnorm_relu(bufA, 64, 2);

  // ---- output MLP: 64->128->256->128 (relu) then 128->1 + sigmoid ----
  launch_gemm(bufA, omW[0], omB[0], bufB, N, 64, 128, 1, stream);
  launch_gemm(bufB, omW[1], omB[1], bufA, N, 128, 256, 1, stream);
  launch_gemm(bufA, omW[2], omB[2], bufB, N, 256, 128, 1, stream);
  hipLaunchKernelGGL(final_out_sigmoid, dim3(cdiv(N, 8)), dim3(T), 0, stream,
                     bufB, omW[3], omB[3], (float*)d_out, N, 128);
}